// SegHead_65343632442119
// MI455X (gfx1250) — compile-verified
//
#include <hip/hip_runtime.h>
#include <math.h>
#include <stdint.h>

// Problem constants (match reference setup_inputs)
#define B_DIM    8
#define C_DIM    128
#define N_DIM    65536
#define S_DIM    100          // NUM_SEGMENTS_PER_PC

// Kernel tiling
#define C_BLK    8            // channels per workgroup: one per wave32
#define N_CHUNKS 16           // split of N; combined via global_atomic_max_num_f32
#define NCHUNK   (N_DIM / N_CHUNKS)     // 4096 points per chunk
#define S_PAD    101          // 0..99 real segments, slot 100 = trash for invalid labels
#define THREADS  256          // 8 waves of 32

__global__ __launch_bounds__(THREADS) void seg_out_init(float* __restrict__ out, int n) {
    int i = blockIdx.x * blockDim.x + threadIdx.x;
    if (i < n) out[i] = -INFINITY;   // segment_max identity (empty segments -> -inf)
}

__global__ __launch_bounds__(THREADS) void seg_max_kernel(
    const float* __restrict__ pf,   // (B, C, N)
    const int*   __restrict__ ids,  // (B, N)
    float*       __restrict__ out)  // (B*S, C)
{
    __shared__ int   lds_labels[NCHUNK];        // 16 KB: staged label chunk
    __shared__ float smax[C_BLK * S_PAD];       // 3.2 KB: per-wave segment maxes

    const int wg    = blockIdx.x;
    const int chunk = wg % N_CHUNKS;
    const int cblk  = (wg / N_CHUNKS) % (C_DIM / C_BLK);
    const int b     = wg / (N_CHUNKS * (C_DIM / C_BLK));

    const int tid  = threadIdx.x;
    const int wave = tid >> 5;      // 0..7 -> channel within block (wave32)
    const int lane = tid & 31;

    const int* lbase = ids + (size_t)b * N_DIM + (size_t)chunk * NCHUNK;

    // --- stage the label chunk into LDS via async DMA (ASYNCcnt-tracked) ---
    // 4096 ints = 16384 B; 256 threads x 4 passes x b128 (16 B/lane)
    {
        const unsigned lds_base = (unsigned)(uintptr_t)&lds_labels[0];
        #pragma unroll
        for (int k = 0; k < 4; ++k) {
            unsigned boff = (unsigned)tid * 16u + (unsigned)k * 4096u;
            unsigned ldsa = lds_base + boff;
            asm volatile("global_load_async_to_lds_b128 %0, %1, %2"
                         :: "v"(ldsa), "v"(boff), "s"(lbase)
                         : "memory");
        }
    }

    // overlap: init LDS accumulators while the async DMA runs
    for (int i = tid; i < C_BLK * S_PAD; i += THREADS) smax[i] = -INFINITY;

    asm volatile("s_wait_asynccnt 0x0" ::: "memory");
    __syncthreads();

    const int c = cblk * C_BLK + wave;
    const float* fbase = pf + ((size_t)b * C_DIM + c) * N_DIM + (size_t)chunk * NCHUNK;
    float* wmax = &smax[wave * S_PAD];

    // each wave streams its chunk: 32 lanes x float4 = 512 B per iter
    const int iters = NCHUNK / (32 * 4);   // 32
    for (int i = 0; i < iters; ++i) {
        const int off = i * 128 + lane * 4;
        // prefetch the feature stream ~4KB ahead (global_prefetch_b8; speculative, fault-safe)
        __builtin_prefetch(fbase + off + 1024, 0, 0);

        float4 f = *reinterpret_cast<const float4*>(fbase + off);
        int4   l = *reinterpret_cast<const int4*>(&lds_labels[off]);   // ds_load_b128

        // clamp invalid labels (e.g. -1) into trash slot 100; atomics run unconditionally
        unsigned sx = ((unsigned)l.x < (unsigned)S_DIM) ? (unsigned)l.x : (unsigned)S_DIM;
        unsigned sy = ((unsigned)l.y < (unsigned)S_DIM) ? (unsigned)l.y : (unsigned)S_DIM;
        unsigned sz = ((unsigned)l.z < (unsigned)S_DIM) ? (unsigned)l.z : (unsigned)S_DIM;
        unsigned sw = ((unsigned)l.w < (unsigned)S_DIM) ? (unsigned)l.w : (unsigned)S_DIM;

        __hip_atomic_fetch_max(&wmax[sx], f.x, __ATOMIC_RELAXED, __HIP_MEMORY_SCOPE_WORKGROUP);
        __hip_atomic_fetch_max(&wmax[sy], f.y, __ATOMIC_RELAXED, __HIP_MEMORY_SCOPE_WORKGROUP);
        __hip_atomic_fetch_max(&wmax[sz], f.z, __ATOMIC_RELAXED, __HIP_MEMORY_SCOPE_WORKGROUP);
        __hip_atomic_fetch_max(&wmax[sw], f.w, __ATOMIC_RELAXED, __HIP_MEMORY_SCOPE_WORKGROUP);
    }
    __syncthreads();

    // flush 8 channels x 100 segments via global_atomic_max_num_f32
    for (int idx = tid; idx < C_BLK * S_DIM; idx += THREADS) {
        const int co = idx & (C_BLK - 1);
        const int s  = idx >> 3;
        const float v = smax[co * S_PAD + s];
        if (v > -INFINITY) {
            float* dst = out + ((size_t)(b * S_DIM + s)) * C_DIM + (size_t)(cblk * C_BLK + co);
            __hip_atomic_fetch_max(dst, v, __ATOMIC_RELAXED, __HIP_MEMORY_SCOPE_AGENT);
        }
    }
}

extern "C" void kernel_launch(void* const* d_in, const int* in_sizes, int n_in,
                              void* d_out, int out_size, void* d_ws, size_t ws_size,
                              hipStream_t stream) {
    const float* pf  = (const float*)d_in[0];   // point_features (B, C, N)
    // d_in[1] = points (B, N, 4) -- unused by the reference computation
    const int*   ids = (const int*)d_in[2];     // box_ids_of_pts (B, N), int32 per harness
    float*       out = (float*)d_out;           // (B*S, C) = (800, 128)

    // 1) initialize output to -inf every call (atomic-max accumulation target)
    {
        int n = out_size;                        // 102400
        int grid = (n + THREADS - 1) / THREADS;
        seg_out_init<<<grid, THREADS, 0, stream>>>(out, n);
    }

    // 2) segmented max scatter
    {
        int grid = B_DIM * (C_DIM / C_BLK) * N_CHUNKS;  // 8 * 16 * 16 = 2048 WGs
        seg_max_kernel<<<grid, THREADS, 0, stream>>>(pf, ids, out);
    }
    (void)in_sizes; (void)n_in; (void)d_ws; (void)ws_size;
}